// TransformerModel_61649960567294
// MI455X (gfx1250) — compile-verified
//
#include <hip/hip_runtime.h>
#include <math.h>

// ---------------------------------------------------------------------------
// Tiny-transformer encoder for MI455X (gfx1250, wave32).
// One wave per (batch, encode) sequence. All GEMMs on V_WMMA_F32_16X16X4_F32
// (native fp32 to match the fp32 reference). LDS staging with ISA fragment
// layouts; s_wait_dscnt for intra-wave LDS ordering.
// ---------------------------------------------------------------------------

typedef __attribute__((ext_vector_type(2))) float v2f;
typedef __attribute__((ext_vector_type(8))) float v8f;

#define SEQ   32
#define DIM   16
#define NWAVE 4           // waves per block (128 threads)
#define XS    17          // padded row stride for [32][16] buffers (bank-conflict free)
#define SS    33          // padded row stride for [32][32] score buffer
#define ATTN_SCALE 0.25f  // 1/sqrt(16)

struct WaveLds {
  float x [SEQ][XS];   // embeddings -> attn context (reused)
  float q [SEQ][XS];   // Q -> hidden (reused)
  float k [SEQ][XS];
  float v [SEQ][XS];
  float sc[SEQ][SS];   // scores -> attn weights
  float red[32];       // mean reduction
};

struct BlockLds {
  float wgt[5][DIM][DIM];  // wq, wk, wv, w1, w2
  float bias[2][DIM];      // b1, b2
  WaveLds wl[NWAVE];
};

// Intra-wave LDS ordering: LDS is HW in-order per wave; the wait + memory
// clobber stops the compiler from reordering/hoisting LDS accesses.
__device__ __forceinline__ void wave_lds_fence() {
  __asm__ volatile("s_wait_dscnt 0x0" ::: "memory");
}

// A fragment (16x4, MxK) OR B fragment when the source is transposed
// (B[k][n] = M[n][k], e.g. x @ W.T and Q @ K.T):
// lane l, component j -> element (i = base + (l&15), k = kc + 2*(l>>4) + j)
__device__ __forceinline__ v2f frag_colk(const float* M, int stride, int base,
                                         int kc, int lane) {
  const int i = base + (lane & 15);
  const int kk = kc + ((lane >> 4) << 1);
  v2f r;
  r.x = M[i * stride + kk];
  r.y = M[i * stride + kk + 1];
  return r;
}

// B fragment (4x16, KxN) from a k-row-major source (B[k][n] = M[k][n], attn @ V):
__device__ __forceinline__ v2f frag_rowk(const float* M, int stride, int base,
                                         int kc, int lane) {
  const int n = base + (lane & 15);
  const int kk = kc + ((lane >> 4) << 1);
  v2f r;
  r.x = M[kk * stride + n];
  r.y = M[(kk + 1) * stride + n];
  return r;
}

#define WMMA4(a, b, c) \
  __builtin_amdgcn_wmma_f32_16x16x4_f32(false, (a), false, (b), (short)0, (c), false, false)

__global__ __launch_bounds__(NWAVE * 32) void encoder_kernel(
    const int* __restrict__ arr1, const int* __restrict__ arr2,
    const float* __restrict__ emb,
    const float* __restrict__ wq, const float* __restrict__ wk,
    const float* __restrict__ wvp,
    const float* __restrict__ w1, const float* __restrict__ b1,
    const float* __restrict__ w2, const float* __restrict__ b2,
    float* __restrict__ out, int B)
{
  __shared__ BlockLds lds;
  const int tid  = threadIdx.x;
  const int lane = tid & 31;
  const int wave = tid >> 5;

  // ---- block-cooperative weight/bias preload (5 KB, L2-resident) ----
  for (int i = tid; i < 5 * 256; i += NWAVE * 32) {
    const int m = i >> 8, r = i & 255;
    const float* p = (m == 0) ? wq : (m == 1) ? wk : (m == 2) ? wvp
                   : (m == 3) ? w1 : w2;
    lds.wgt[m][r >> 4][r & 15] = p[r];
  }
  if (tid < DIM) {
    lds.bias[0][tid] = b1[tid];
    lds.bias[1][tid] = b2[tid];
  }
  __syncthreads();

  long job = (long)blockIdx.x * NWAVE + wave;
  const long njobs = 2L * (long)B;
  if (job >= njobs) job = njobs - 1;        // uniform clamp: no divergent waves
  const int enc = (int)(job >= (long)B);
  const int bidx = (int)(job - (long)enc * B);
  const int* arr = enc ? arr2 : arr1;
  WaveLds& L = lds.wl[wave];

  const int nn = lane & 15;                 // column owned by this lane (D/C frags)
  const int rhalf = (lane >> 4) << 3;       // row-half offset for D frags

  // ---- 1. embedding gather: lane l loads token row s=l (64B = one float4x4) ----
  {
    const int tok = arr[bidx * SEQ + lane];
    const float4* e4 = (const float4*)(emb + (long)tok * DIM);
    const float4 a0 = e4[0], a1 = e4[1], a2 = e4[2], a3 = e4[3];
    float* row = &L.x[lane][0];
    row[0]  = a0.x; row[1]  = a0.y; row[2]  = a0.z; row[3]  = a0.w;
    row[4]  = a1.x; row[5]  = a1.y; row[6]  = a1.z; row[7]  = a1.w;
    row[8]  = a2.x; row[9]  = a2.y; row[10] = a2.z; row[11] = a2.w;
    row[12] = a3.x; row[13] = a3.y; row[14] = a3.z; row[15] = a3.w;
  }
  wave_lds_fence();

  // ---- 2. Q/K/V projections: [16x16] tile per row-half, K=16 -> 4 WMMAs ----
#pragma unroll
  for (int t = 0; t < 2; ++t) {
    v8f qa = {}, ka = {}, va = {};
#pragma unroll
    for (int kc = 0; kc < DIM; kc += 4) {
      const v2f a  = frag_colk(&L.x[0][0], XS, t * 16, kc, lane);
      const v2f bq = frag_colk(&lds.wgt[0][0][0], DIM, 0, kc, lane);
      const v2f bk = frag_colk(&lds.wgt[1][0][0], DIM, 0, kc, lane);
      const v2f bv = frag_colk(&lds.wgt[2][0][0], DIM, 0, kc, lane);
      qa = WMMA4(a, bq, qa);
      ka = WMMA4(a, bk, ka);
      va = WMMA4(a, bv, va);
    }
    const int r0 = t * 16 + rhalf;
#pragma unroll
    for (int r = 0; r < 8; ++r) {
      L.q[r0 + r][nn] = qa[r];
      L.k[r0 + r][nn] = ka[r];
      L.v[r0 + r][nn] = va[r];
    }
  }
  wave_lds_fence();

  // ---- 3. scores = (Q @ K.T) * scale : four 16x16 tiles ----
#pragma unroll
  for (int ti = 0; ti < 2; ++ti) {
#pragma unroll
    for (int tj = 0; tj < 2; ++tj) {
      v8f s = {};
#pragma unroll
      for (int kc = 0; kc < DIM; kc += 4) {
        const v2f a  = frag_colk(&L.q[0][0], XS, ti * 16, kc, lane);
        const v2f bb = frag_colk(&L.k[0][0], XS, tj * 16, kc, lane);  // K^T
        s = WMMA4(a, bb, s);
      }
      const int r0 = ti * 16 + rhalf;
      const int c0 = tj * 16 + nn;
#pragma unroll
      for (int r = 0; r < 8; ++r) L.sc[r0 + r][c0] = s[r] * ATTN_SCALE;
    }
  }
  wave_lds_fence();

  // ---- 4. softmax: lane l owns row s=l (32 elements, stride-33 = conflict-free) ----
  {
    float row[SEQ];
#pragma unroll
    for (int j = 0; j < SEQ; ++j) row[j] = L.sc[lane][j];
    float mx = row[0];
#pragma unroll
    for (int j = 1; j < SEQ; ++j) mx = fmaxf(mx, row[j]);
    float sum = 0.f;
#pragma unroll
    for (int j = 0; j < SEQ; ++j) { row[j] = __expf(row[j] - mx); sum += row[j]; }
    const float inv = 1.f / sum;
#pragma unroll
    for (int j = 0; j < SEQ; ++j) L.sc[lane][j] = row[j] * inv;
  }
  wave_lds_fence();

  // ---- 5. ctx = attn @ V : K=32 -> 8 WMMAs per tile ----
#pragma unroll
  for (int t = 0; t < 2; ++t) {
    v8f c = {};
#pragma unroll
    for (int kc = 0; kc < SEQ; kc += 4) {
      const v2f a  = frag_colk(&L.sc[0][0], SS, t * 16, kc, lane);
      const v2f bb = frag_rowk(&L.v[0][0], XS, 0, kc, lane);
      c = WMMA4(a, bb, c);
    }
    const int r0 = t * 16 + rhalf;
#pragma unroll
    for (int r = 0; r < 8; ++r) L.x[r0 + r][nn] = c[r];   // reuse x as ctx
  }
  wave_lds_fence();

  // ---- 6. FFN layer 1: relu(ctx @ w1.T + b1) -> reuse q buffer ----
  const float bias1 = lds.bias[0][nn];
  const float bias2 = lds.bias[1][nn];
#pragma unroll
  for (int t = 0; t < 2; ++t) {
    v8f h = {};
#pragma unroll
    for (int kc = 0; kc < DIM; kc += 4) {
      const v2f a  = frag_colk(&L.x[0][0], XS, t * 16, kc, lane);
      const v2f bb = frag_colk(&lds.wgt[3][0][0], DIM, 0, kc, lane);
      h = WMMA4(a, bb, h);
    }
    const int r0 = t * 16 + rhalf;
#pragma unroll
    for (int r = 0; r < 8; ++r) L.q[r0 + r][nn] = fmaxf(h[r] + bias1, 0.f);
  }
  wave_lds_fence();

  // ---- 7. FFN layer 2 + mean over S (row-sum falls out of D-frag layout) ----
  float partial = 0.f;
#pragma unroll
  for (int t = 0; t < 2; ++t) {
    v8f y = {};
#pragma unroll
    for (int kc = 0; kc < DIM; kc += 4) {
      const v2f a  = frag_colk(&L.q[0][0], XS, t * 16, kc, lane);
      const v2f bb = frag_colk(&lds.wgt[4][0][0], DIM, 0, kc, lane);
      y = WMMA4(a, bb, y);
    }
#pragma unroll
    for (int r = 0; r < 8; ++r) partial += y[r] + bias2;  // 16 rows per lane total
  }
  L.red[lane] = partial;
  wave_lds_fence();

  if (lane < 16) {
    const float res = (L.red[lane] + L.red[lane + 16]) * (1.0f / 32.0f);
    out[((long)enc * B + bidx) * DIM + lane] = res;
  }
}

extern "C" void kernel_launch(void* const* d_in, const int* in_sizes, int n_in,
                              void* d_out, int out_size, void* d_ws, size_t ws_size,
                              hipStream_t stream) {
  const int*   arr1 = (const int*)  d_in[0];
  const int*   arr2 = (const int*)  d_in[1];
  const float* emb  = (const float*)d_in[2];
  const float* wq   = (const float*)d_in[3];
  const float* wk   = (const float*)d_in[4];
  const float* wv   = (const float*)d_in[5];
  const float* w1   = (const float*)d_in[6];
  const float* b1   = (const float*)d_in[7];
  const float* w2   = (const float*)d_in[8];
  const float* b2   = (const float*)d_in[9];
  float* out = (float*)d_out;

  const int  B     = in_sizes[0] / SEQ;      // 65536
  const long jobs  = 2L * (long)B;           // two encodes
  const int  blocks = (int)((jobs + NWAVE - 1) / NWAVE);

  encoder_kernel<<<blocks, NWAVE * 32, 0, stream>>>(
      arr1, arr2, emb, wq, wk, wv, w1, b1, w2, b2, out, B);
}